// MaskBalanceBCELoss_60301340836062
// MI455X (gfx1250) — compile-verified
//
#include <hip/hip_runtime.h>
#include <hip/hip_bf16.h>
#include <stdint.h>

// ---------------- configuration ----------------
#define NBINS      4096
#define BIN_SHIFT  19            // float bits >> 19 -> 12-bit monotone bin (loss in [0,100])
#define FXSCALE_D  1073741824.0  // 2^30 fixed-point scale for wave-level scalar sums
#define FXBIN_F    16777216.0f   // 2^24 fixed-point scale for per-element bin sums (loss<=100 fits u32)
#define FXBIN_D    16777216.0
#define LN2_F      0.69314718056f
#define P1_BLOCKS  2048
#define P1_THREADS 256

// workspace layout (bytes)
//   [0,      16384)  u32  binCnt[4096]
//   [16384,  49152)  u64  binFx [4096]      (scale 2^24)
//   [49152,  49160)  u64  posFx             (scale 2^30)
//   [49160,  49168)  u64  negFx             (scale 2^30)
//   [49168,  49172)  u32  posCnt
//   [49172,  49176)  u32  negCnt
#define WS_WORDS   12294         // 49176 / 4

typedef float v2f __attribute__((ext_vector_type(2)));
typedef float v8f __attribute__((ext_vector_type(8)));

__device__ __forceinline__ v8f wmma_rowsum(v2f a, v2f b, v8f c) {
  // D = A(16x4,f32) * B(4x16,f32) + C ; with B == ones this row-sums A into every D column.
  return __builtin_amdgcn_wmma_f32_16x16x4_f32(
      /*neg_a=*/false, a, /*neg_b=*/false, b,
      /*c_mod=*/(short)0, c, /*reuse_a=*/false, /*reuse_b=*/false);
}

__device__ __forceinline__ unsigned bin_of(float v) {
  return (__float_as_uint(v) >> BIN_SHIFT) & (NBINS - 1u);
}

// BCE loss term: min(-ln(x), 100), x >= 1e-4 (no denorms) ->
// single raw v_log_f32 (log2) + one multiply (by -ln2) + min.
__device__ __forceinline__ float loss_of(float x) {
  return fminf(__builtin_amdgcn_logf(x) * (-LN2_F), 100.0f);
}

__global__ void __launch_bounds__(256)
bce_zero_ws(unsigned* ws, unsigned words) {
  unsigned i = blockIdx.x * blockDim.x + threadIdx.x;
  if (i < words) ws[i] = 0u;
}

__global__ void __launch_bounds__(P1_THREADS)
bce_pass1(const float* __restrict__ pred, const int* __restrict__ gt,
          const int* __restrict__ mask, unsigned n,
          unsigned* __restrict__ gBinCnt, unsigned long long* __restrict__ gBinFx,
          unsigned long long* __restrict__ gPosFx, unsigned long long* __restrict__ gNegFx,
          unsigned* __restrict__ gPosCnt, unsigned* __restrict__ gNegCnt)
{
  __shared__ unsigned           sCnt[NBINS];
  __shared__ unsigned long long sFx [NBINS];
  for (int i = threadIdx.x; i < NBINS; i += P1_THREADS) { sCnt[i] = 0u; sFx[i] = 0ull; }
  __syncthreads();

  const unsigned lane   = threadIdx.x & 31u;
  const unsigned chunks = n >> 2;                              // float4 chunks (fits 32-bit)
  unsigned base = blockIdx.x * P1_THREADS + (threadIdx.x & ~31u);
  const unsigned stride = gridDim.x * P1_THREADS;

  const float4* __restrict__ p4 = (const float4*)pred;
  const int4*   __restrict__ g4 = (const int4*)gt;
  const int4*   __restrict__ m4 = (const int4*)mask;

  v8f accPL = {0.f,0.f,0.f,0.f,0.f,0.f,0.f,0.f};
  v8f accNL = accPL, accPC = accPL, accNC = accPL;
  const v2f bones = {1.0f, 1.0f};

  // Branchless per-element accumulation, single transcendental per element;
  // only the histogram atomic is conditional (integer predicate).
  auto elem = [&](float p, int g, int m, float& pl, float& nl, float& pc, float& nc) {
    const bool  gpos = (g != 0);
    const bool  neg  = (!gpos) & (m != 0);
    const float mf   = (m != 0) ? 1.0f : 0.0f;
    const float gf   = gpos ? 1.0f : 0.0f;
    const float posf = gf * mf;
    const float negf = mf - posf;                 // (1-gf)*mf
    const float x    = gpos ? p : (1.0f - p);     // the only log the loss needs
    const float loss = loss_of(x);
    pl = fmaf(posf, loss, pl);  pc += posf;
    nl = fmaf(negf, loss, nl);  nc += negf;
    if (neg) {
      const unsigned b = bin_of(loss);
      atomicAdd(&sCnt[b], 1u);
      // single v_cvt_u32_f32, zero-extended into the u64 atomic (scale 2^24)
      atomicAdd(&sFx[b], (unsigned long long)((unsigned)(loss * FXBIN_F)));
    }
  };

  while (base < chunks) {               // base is wave-uniform -> EXEC full at WMMA
    const unsigned idx = base + lane;
    float plA=0.f, plB=0.f, nlA=0.f, nlB=0.f, pcA=0.f, pcB=0.f, ncA=0.f, ncB=0.f;
    if (idx < chunks) {                 // out-of-range lanes contribute exact zeros
      const float4 p = p4[idx];
      const int4   g = g4[idx];
      const int4   m = m4[idx];
      elem(p.x, g.x, m.x, plA, nlA, pcA, ncA);
      elem(p.y, g.y, m.y, plA, nlA, pcA, ncA);
      elem(p.z, g.z, m.z, plB, nlB, pcB, ncB);
      elem(p.w, g.w, m.w, plB, nlB, pcB, ncB);
    }
    v2f aPL = {plA, plB}; accPL = wmma_rowsum(aPL, bones, accPL);
    v2f aNL = {nlA, nlB}; accNL = wmma_rowsum(aNL, bones, accNL);
    v2f aPC = {pcA, pcB}; accPC = wmma_rowsum(aPC, bones, accPC);
    v2f aNC = {ncA, ncB}; accNC = wmma_rowsum(aNC, bones, accNC);
    base += stride;
  }

  // D rows 0-7 live in lanes 0-15, rows 8-15 in lanes 16-31; columns identical (ones-B).
  float sPL = 0.f, sNL = 0.f, sPC = 0.f, sNC = 0.f;
  for (int i = 0; i < 8; ++i) { sPL += accPL[i]; sNL += accNL[i]; sPC += accPC[i]; sNC += accNC[i]; }
  sPL += __shfl_xor(sPL, 16); sNL += __shfl_xor(sNL, 16);
  sPC += __shfl_xor(sPC, 16); sNC += __shfl_xor(sNC, 16);

  if (lane == 0) {   // per-wave scalar flush, deterministic u64 fixed-point atomics
    atomicAdd(gPosFx, (unsigned long long)((double)sPL * FXSCALE_D));
    atomicAdd(gNegFx, (unsigned long long)((double)sNL * FXSCALE_D));
    atomicAdd(gPosCnt, (unsigned)(sPC + 0.5f));
    atomicAdd(gNegCnt, (unsigned)(sNC + 0.5f));
  }

  __syncthreads();
  for (int i = threadIdx.x; i < NBINS; i += P1_THREADS) {
    const unsigned c = sCnt[i];
    if (c) { atomicAdd(&gBinCnt[i], c); atomicAdd(&gBinFx[i], sFx[i]); }
  }

  // scalar tail (n % 4), single thread, deterministic
  if (blockIdx.x == 0 && threadIdx.x == 0) {
    for (unsigned i = chunks << 2; i < n; ++i) {
      const float p = pred[i]; const int g = gt[i]; const int m = mask[i];
      if (m != 0) {
        const float loss = loss_of((g != 0) ? p : (1.0f - p));
        if (g != 0) {
          atomicAdd(gPosFx, (unsigned long long)((double)loss * FXSCALE_D));
          atomicAdd(gPosCnt, 1u);
        } else {
          atomicAdd(gNegFx, (unsigned long long)((double)loss * FXSCALE_D));
          atomicAdd(gNegCnt, 1u);
          const unsigned b = bin_of(loss);
          atomicAdd(&gBinCnt[b], 1u);
          atomicAdd(&gBinFx[b], (unsigned long long)((unsigned)(loss * FXBIN_F)));
        }
      }
    }
  }
}

__global__ void __launch_bounds__(256)
bce_pass2(const unsigned* __restrict__ gBinCnt, const unsigned long long* __restrict__ gBinFx,
          const unsigned long long* __restrict__ gPosFx, const unsigned long long* __restrict__ gNegFx,
          const unsigned* __restrict__ gPosCnt, const unsigned* __restrict__ gNegCnt,
          float* __restrict__ out)
{
  __shared__ unsigned long long grpCnt[256];
  __shared__ unsigned long long grpFx [256];
  const int t = threadIdx.x;
  unsigned long long c = 0ull, f = 0ull;
  for (int b = t * 16; b < t * 16 + 16; ++b) { c += gBinCnt[b]; f += gBinFx[b]; }
  grpCnt[t] = c; grpFx[t] = f;
  __syncthreads();

  if (t == 0) {
    const double posCnt  = (double)(*gPosCnt);
    const double negCnt  = (double)(*gNegCnt);
    const double posLoss = (double)(*gPosFx) / FXSCALE_D;
    const double negTot  = (double)(*gNegFx) / FXSCALE_D;
    const double kd      = fmin(negCnt, 3.0 * posCnt);   // negative_count (integer-valued)

    double negLoss;
    if (kd >= negCnt) {
      negLoss = negTot;                                  // keep all negatives (expected case)
    } else {
      const long long k = (long long)(kd + 0.5);
      negLoss = 0.0;
      if (k > 0) {
        long long cum = 0;
        unsigned long long fxAbove = 0ull;               // scale 2^24
        int g;
        for (g = 255; g >= 0; --g) {
          if (cum + (long long)grpCnt[g] >= k) break;
          cum += (long long)grpCnt[g];
          fxAbove += grpFx[g];
        }
        if (g >= 0) {
          for (int b = g * 16 + 15; b >= g * 16 && cum < k; --b) {
            const unsigned bc = gBinCnt[b];
            if (!bc) continue;
            if (cum + (long long)bc <= k) {
              cum += (long long)bc;
              fxAbove += gBinFx[b];
            } else {
              const long long rem = k - cum;
              const double avgFx  = (double)gBinFx[b] / (double)bc;
              negLoss += (double)rem * avgFx / FXBIN_D;  // boundary bin: within-bin average
              cum = k;
            }
          }
        }
        negLoss += (double)fxAbove / FXBIN_D;
      }
    }
    const double denom = posCnt + kd + 1e-6;
    out[0] = (float)((posLoss + negLoss) / denom);
  }
}

extern "C" void kernel_launch(void* const* d_in, const int* in_sizes, int n_in,
                              void* d_out, int out_size, void* d_ws, size_t ws_size,
                              hipStream_t stream) {
  const float* pred = (const float*)d_in[0];
  const int*   gt   = (const int*)d_in[1];
  const int*   mask = (const int*)d_in[2];
  const unsigned n  = (unsigned)in_sizes[0];

  char* ws = (char*)d_ws;
  unsigned*           gBinCnt = (unsigned*)(ws + 0);
  unsigned long long* gBinFx  = (unsigned long long*)(ws + 16384);
  unsigned long long* gPosFx  = (unsigned long long*)(ws + 49152);
  unsigned long long* gNegFx  = (unsigned long long*)(ws + 49160);
  unsigned*           gPosCnt = (unsigned*)(ws + 49168);
  unsigned*           gNegCnt = (unsigned*)(ws + 49172);

  bce_zero_ws<<<(WS_WORDS + 255) / 256, 256, 0, stream>>>((unsigned*)ws, WS_WORDS);

  unsigned chunks = n >> 2;
  unsigned blocks = (chunks + P1_THREADS - 1) / P1_THREADS;
  if (blocks > P1_BLOCKS) blocks = P1_BLOCKS;
  if (blocks == 0) blocks = 1;
  bce_pass1<<<blocks, P1_THREADS, 0, stream>>>(pred, gt, mask, n,
                                               gBinCnt, gBinFx, gPosFx, gNegFx,
                                               gPosCnt, gNegCnt);

  bce_pass2<<<1, 256, 0, stream>>>(gBinCnt, gBinFx, gPosFx, gNegFx,
                                   gPosCnt, gNegCnt, (float*)d_out);
}